// SparseMatchingStereoModel_4277787427012
// MI455X (gfx1250) — compile-verified
//
#include <hip/hip_runtime.h>
#include <math.h>

#define B_     32
#define H_     256
#define W_     512
#define HW_    (H_*W_)
#define C_     768
#define HF_    16
#define WF_    32
#define KC_    2048      // K_CAND
#define MAXKP_ 512
#define CAP_   4096      // candidate buffer capacity (>= KC_, slack for ties)
#define THRESH_ 0.1f
#define R2_     9.0f     // NMS_RADIUS^2

typedef float v2f __attribute__((ext_vector_type(2)));
typedef float v8f __attribute__((ext_vector_type(8)));

// ---------------------------------------------------------------- init state
__global__ void init_kernel(unsigned* prefix, unsigned* kremain, unsigned* cnt,
                            unsigned long long* cbuf) {
  int i = blockIdx.x * blockDim.x + threadIdx.x;
  if (i < B_ * CAP_) cbuf[i] = ~0ull;               // sorts to the end
  if (i < B_) { prefix[i] = 0u; kremain[i] = KC_; cnt[i] = 0u; }
}

// ---------------------------------------------------- radix select (1 byte/pass)
// Non-negative floats compare identically to their bit patterns as uints.
__global__ void radix_pass_kernel(const float* __restrict__ img,
                                  const float* __restrict__ mask,
                                  unsigned* prefix, unsigned* kremain, int pass) {
  __shared__ unsigned hist[256];
  int b = blockIdx.x, tid = threadIdx.x;
  if (tid < 256) hist[tid] = 0u;
  __syncthreads();
  int shift = 24 - 8 * pass;
  unsigned prefmask = (pass == 0) ? 0u : (0xFFFFFFFFu << (32 - 8 * pass));
  unsigned pref = prefix[b];
  const float* ib = img  + (size_t)b * HW_;
  const float* mb = mask + (size_t)b * HW_;
  for (int i = tid; i < HW_; i += blockDim.x) {
    unsigned bits = __float_as_uint(ib[i] * mb[i]);
    if ((bits & prefmask) == pref) atomicAdd(&hist[(bits >> shift) & 255u], 1u);
  }
  __syncthreads();
  if (tid == 0) {
    unsigned kr = kremain[b], acc = 0u;
    int t = 255;
    for (; t > 0; --t) {                 // find byte bin holding the k-th value
      if (acc + hist[t] >= kr) break;
      acc += hist[t];
    }
    kremain[b] = kr - acc;
    prefix[b]  = pref | ((unsigned)t << shift);
  }
}

// ----------------------------------------------------------- collect candidates
__global__ void collect_kernel(const float* __restrict__ img,
                               const float* __restrict__ mask,
                               const unsigned* prefix, unsigned* cnt,
                               unsigned long long* cbuf) {
  int b = blockIdx.x, tid = threadIdx.x;
  unsigned T = prefix[b];                 // exact bits of 2048th largest value
  const float* ib = img  + (size_t)b * HW_;
  const float* mb = mask + (size_t)b * HW_;
  for (int i = tid; i < HW_; i += blockDim.x) {
    unsigned bits = __float_as_uint(ib[i] * mb[i]);
    if (bits >= T) {
      unsigned pos = atomicAdd(&cnt[b], 1u);
      if (pos < CAP_)   // key: ascending sort => value desc, index asc (top_k order)
        cbuf[(size_t)b * CAP_ + pos] =
            (((unsigned long long)(~bits)) << 32) | (unsigned)i;
    }
  }
}

// ------------------------------------------------- per-batch bitonic sort (LDS)
__global__ void sort_kernel(const unsigned long long* __restrict__ cbuf,
                            float* cvals, int* cidx) {
  __shared__ unsigned long long s[CAP_];            // 32 KB
  int b = blockIdx.x, tid = threadIdx.x;
  for (int i = tid; i < CAP_; i += blockDim.x) s[i] = cbuf[(size_t)b * CAP_ + i];
  __syncthreads();
  for (unsigned k = 2; k <= CAP_; k <<= 1) {
    for (unsigned j = k >> 1; j > 0; j >>= 1) {
      for (unsigned i = tid; i < CAP_; i += blockDim.x) {
        unsigned ixj = i ^ j;
        if (ixj > i) {
          unsigned long long a = s[i], c = s[ixj];
          bool up = ((i & k) == 0u);
          if ((a > c) == up) { s[i] = c; s[ixj] = a; }
        }
      }
      __syncthreads();
    }
  }
  for (int i = tid; i < KC_; i += blockDim.x) {
    unsigned long long key = s[i];
    cvals[b * KC_ + i] = __uint_as_float(~(unsigned)(key >> 32));
    cidx [b * KC_ + i] = (int)(unsigned)key;
  }
}

// --------------------------------------------------------- WMMA-tiled greedy NMS
// One wave32 per batch. dist^2(i,j) = A_i . B_j with
//   A_i = [x, y, x^2+y^2, 1],  B_j = [-2x, -2y, 1, x^2+y^2]  (K = 4, f32, exact)
__global__ void __launch_bounds__(32) nms_kernel(const float* __restrict__ cvals,
                                                 const int* __restrict__ cidx,
                                                 unsigned* keep) {
  __shared__ float sx[KC_], sy[KC_];                // 16 KB
  __shared__ unsigned skeep[KC_ / 32];              // 64 words: valid -> final keep
  int b = blockIdx.x, lane = threadIdx.x;
  for (int i = lane; i < KC_; i += 32) {
    int id = cidx[b * KC_ + i];
    sx[i] = (float)(id & (W_ - 1));
    sy[i] = (float)(id >> 9);
  }
  for (int w = lane; w < KC_ / 32; w += 32) {
    unsigned bits = 0u;
    for (int t = 0; t < 32; ++t)
      if (cvals[b * KC_ + w * 32 + t] > THRESH_) bits |= (1u << t);
    skeep[w] = bits;                                // initially = valid
  }
  __syncthreads();

  const int hv = lane >> 4;                         // wave half
  const int mrow = lane & 15;                       // A row / B col index
  for (int I = 0; I < KC_ / 16; ++I) {
    int i0 = I * 16;
    float ax = sx[i0 + mrow], ay = sy[i0 + mrow];
    v2f A;                                          // A 16x4: half0 -> K0,1; half1 -> K2,3
    if (hv == 0) { A[0] = ax;               A[1] = ay;   }
    else         { A[0] = ax * ax + ay * ay; A[1] = 1.0f; }

    unsigned supLocal = 0u;                         // bit m: suppressed by kept j<i0
    for (int J = 0; J < I; ++J) {
      int j0 = J * 16;
      float bx = sx[j0 + mrow], by = sy[j0 + mrow];
      v2f Bv;                                       // B 4x16 column
      if (hv == 0) { Bv[0] = -2.0f * bx; Bv[1] = -2.0f * by;       }
      else         { Bv[0] = 1.0f;       Bv[1] = bx * bx + by * by; }
      v8f Cz = {};
      v8f D = __builtin_amdgcn_wmma_f32_16x16x4_f32(
          false, A, false, Bv, (short)0, Cz, false, false);
      int j = j0 + mrow;
      bool jkeep = (skeep[j >> 5] >> (j & 31)) & 1u;   // earlier tiles finalized
      if (jkeep) {
        #pragma unroll
        for (int r = 0; r < 8; ++r)                 // D: M = r + 8*half, N = lane&15
          if (D[r] < R2_) supLocal |= (1u << (r + 8 * hv));
      }
    }
    // OR-reduce suppression bits across the wave
    for (int off = 16; off > 0; off >>= 1)
      supLocal |= __shfl_xor(supLocal, off, 32);
    unsigned sup = supLocal & 0xFFFFu;
    unsigned validTile = (skeep[i0 >> 5] >> (i0 & 31)) & 0xFFFFu;
    // serial greedy resolve inside the diagonal tile (redundant on all lanes)
    for (int a = 0; a < 16; ++a) {
      if (((validTile >> a) & 1u) && !((sup >> a) & 1u)) {
        float xa = sx[i0 + a], ya = sy[i0 + a];
        for (int bq = a + 1; bq < 16; ++bq) {
          float dx = sx[i0 + bq] - xa, dy = sy[i0 + bq] - ya;
          if (dx * dx + dy * dy < R2_) sup |= (1u << bq);
        }
      }
    }
    unsigned keepTile = validTile & ~sup;
    if (lane == 0) {
      int wdx = i0 >> 5, sh = i0 & 31;
      skeep[wdx] = (skeep[wdx] & ~(0xFFFFu << sh)) | (keepTile << sh);
    }
    __syncthreads();
  }
  for (int w = lane; w < KC_ / 32; w += 32) keep[b * (KC_ / 32) + w] = skeep[w];
}

// ----------------------------------------------------- stable compaction to 512
__global__ void select_kernel(const unsigned* __restrict__ keep, int* sel,
                              unsigned* nkeep) {
  int b = blockIdx.x;
  if (threadIdx.x != 0) return;
  int rank = 0;
  for (int i = 0; i < KC_; ++i) {
    if ((keep[b * (KC_ / 32) + (i >> 5)] >> (i & 31)) & 1u) {
      if (rank < MAXKP_) sel[b * MAXKP_ + rank] = i;
      ++rank;
    }
  }
  for (int r = (rank < MAXKP_ ? rank : MAXKP_); r < MAXKP_; ++r)
    sel[b * MAXKP_ + r] = -1;   // slot beyond n_keep: kp=(0,0), sc=0, desc=fm[:,0,0]
  nkeep[b] = (unsigned)rank;
}

// ------------------------------------------- keypoints, scores, bilinear descs
__global__ void output_kernel(const float* __restrict__ fm,
                              const float* __restrict__ cvals,
                              const int* __restrict__ cidx,
                              const int* __restrict__ sel,
                              float* kp_out, float* sc_out, float* desc_out) {
  int blk = blockIdx.x;
  int b = blk >> 9;            // MAXKP_ == 512
  int k = blk & (MAXKP_ - 1);
  int tid = threadIdx.x;
  int s = sel[b * MAXKP_ + k];
  float x = 0.f, y = 0.f, scv = 0.f;
  if (s >= 0) {
    int id = cidx[b * KC_ + s];
    x = (float)(id & (W_ - 1));
    y = (float)(id >> 9);
    scv = cvals[b * KC_ + s];
  }
  if (tid == 0) {
    kp_out[(b * MAXKP_ + k) * 2 + 0] = x;
    kp_out[(b * MAXKP_ + k) * 2 + 1] = y;
    sc_out[b * MAXKP_ + k] = scv;
  }
  float fx = x * ((float)(WF_ - 1) / (float)W_);
  float fy = y * ((float)(HF_ - 1) / (float)H_);
  float x0f = floorf(fx), y0f = floorf(fy);
  float wx = fx - x0f, wy = fy - y0f;
  int x0 = (int)fminf(fmaxf(x0f,       0.f), (float)(WF_ - 1));
  int x1 = (int)fminf(fmaxf(x0f + 1.f, 0.f), (float)(WF_ - 1));
  int y0 = (int)fminf(fmaxf(y0f,       0.f), (float)(HF_ - 1));
  int y1 = (int)fminf(fmaxf(y0f + 1.f, 0.f), (float)(HF_ - 1));
  float w00 = (1.f - wx) * (1.f - wy), w01 = wx * (1.f - wy);
  float w10 = (1.f - wx) * wy,         w11 = wx * wy;
  const float* fb = fm + (size_t)b * C_ * HF_ * WF_;
  for (int c = tid; c < C_; c += blockDim.x) {
    const float* fc = fb + c * (HF_ * WF_);
    float d = fc[y0 * WF_ + x0] * w00 + fc[y0 * WF_ + x1] * w01 +
              fc[y1 * WF_ + x0] * w10 + fc[y1 * WF_ + x1] * w11;
    desc_out[((size_t)(b * MAXKP_ + k)) * C_ + c] = d;
  }
}

// --------------------------------------------------------------------- launch
extern "C" void kernel_launch(void* const* d_in, const int* in_sizes, int n_in,
                              void* d_out, int out_size, void* d_ws, size_t ws_size,
                              hipStream_t stream) {
  const float* img  = (const float*)d_in[0];   // (32,1,256,512)
  const float* mask = (const float*)d_in[1];   // (32,1,256,512)
  const float* fm   = (const float*)d_in[2];   // (32,768,16,32)

  float* out      = (float*)d_out;
  float* kp_out   = out;                         // (32,512,2)
  float* sc_out   = out + B_ * MAXKP_ * 2;       // (32,512)
  float* desc_out = sc_out + B_ * MAXKP_;        // (32,512,768)

  unsigned char* ws = (unsigned char*)d_ws;
  unsigned* prefix  = (unsigned*)ws;  ws += B_ * 4;
  unsigned* kremain = (unsigned*)ws;  ws += B_ * 4;
  unsigned* cnt     = (unsigned*)ws;  ws += B_ * 4;
  unsigned* nkeep   = (unsigned*)ws;  ws += B_ * 4;      // 512 B so far (8B aligned)
  unsigned long long* cbuf = (unsigned long long*)ws; ws += (size_t)B_ * CAP_ * 8;
  float* cvals = (float*)ws;          ws += (size_t)B_ * KC_ * 4;
  int*   cidx  = (int*)ws;            ws += (size_t)B_ * KC_ * 4;
  unsigned* keep = (unsigned*)ws;     ws += B_ * (KC_ / 32) * 4;
  int*   sel   = (int*)ws;            ws += B_ * MAXKP_ * 4;   // ~1.6 MB total

  init_kernel<<<(B_ * CAP_ + 255) / 256, 256, 0, stream>>>(prefix, kremain, cnt, cbuf);
  for (int p = 0; p < 4; ++p)
    radix_pass_kernel<<<B_, 256, 0, stream>>>(img, mask, prefix, kremain, p);
  collect_kernel<<<B_, 256, 0, stream>>>(img, mask, prefix, cnt, cbuf);
  sort_kernel<<<B_, 512, 0, stream>>>(cbuf, cvals, cidx);
  nms_kernel<<<B_, 32, 0, stream>>>(cvals, cidx, keep);
  select_kernel<<<B_, 32, 0, stream>>>(keep, sel, nkeep);
  output_kernel<<<B_ * MAXKP_, 256, 0, stream>>>(fm, cvals, cidx, sel,
                                                 kp_out, sc_out, desc_out);
}